// LovaszBCEWithLogitsLoss_2774548873765
// MI455X (gfx1250) — compile-verified
//
#include <hip/hip_runtime.h>
#include <hip/hip_bf16.h>

// ---------------------------------------------------------------------------
// Lovasz hinge + BCEWithLogits for B=128 images of 512x512 f32.
//
// MI455X reasoning:
//  * 256 MB mandatory input reads @ 23.3 TB/s  => ~11 us streaming floor.
//  * A real per-image sort (radix) would add ~2 GB traffic => avoid it.
//  * Tie-group telescoping of the Lovasz sum means only per-distinct-error
//    aggregates are needed: (count, pos-count, sum relu(e)).  A fine monotone
//    bucketing (16384 buckets over e in [-8,8], ~1e-3 wide) makes this a
//    single histogram pass + a 16K bucket scan per image.
//  * 320 KB LDS per WGP (CDNA5) lets the whole 3-field per-image histogram
//    live in LDS: u64 {pos:32|cnt:32} + u64 fixed-point sum-relu  = 256 KB.
//    Integer LDS atomics (ds_add_u64) => bit-deterministic across replays.
//  * Cross-wave prefix scan of the 32 wave totals (cnt & pos packed as 4
//    columns of a 16x16 matrix) is done on the tensor pipe:
//    P = T(lower-tri ones) x S  via 4 chained v_wmma_f32_16x16x4_f32.
//    Counts < 2^24 are exact in f32.
// ---------------------------------------------------------------------------

#define B_IMG   128
#define P_IMG   (512 * 512)          // 262144 pixels per image
#define N_TOT   (128LL * 512 * 512)  // 33554432 total elements
#define NB      16384                // buckets per image
#define THREADS 1024                 // 32 wave32 waves
#define EMAX    8.0f
#define BPU     1024.0f              // NB / (2*EMAX) buckets per unit of e
#define FXS     16777216.0f          // 2^24 fixed-point scale for sum relu(e)
#define FXS_INV (1.0f / 16777216.0f)

typedef __attribute__((ext_vector_type(2))) float v2f;
typedef __attribute__((ext_vector_type(8))) float v8f;

// J(r, p) = jaccard after rank r (0-based) with p positives seen, given gts
// total positives:  1 - (gts - p) / (gts + (r+1) - p).   J(-1, 0) = 0.
__device__ __forceinline__ float jacc(float rp1, float p, float gts) {
    // rp1 = r + 1  (>= 1 when called)
    return 1.0f - (gts - p) / (gts + rp1 - p);
}

__global__ __launch_bounds__(THREADS, 1)
void lovasz_bce_kernel(const float* __restrict__ pred,
                       const float* __restrict__ targ,
                       float* __restrict__ ws /* [0..127]=lovasz, [128..255]=bce */) {
    // -------- LDS (256.5 KB of the WGP's 320 KB) --------
    __shared__ unsigned long long s_cp[NB];   // {pos:32 | cnt:32} per bucket
    __shared__ unsigned long long s_fx[NB];   // sum relu(e) * 2^24 per bucket
    __shared__ float s_waveC[32], s_waveQ[32];  // per-wave totals (cnt, pos)
    __shared__ float s_inclC[32], s_inclQ[32];  // inclusive cross-wave prefix

    const int tid  = threadIdx.x;
    const int lane = tid & 31;
    const int wave = tid >> 5;            // 0..31
    const int img  = blockIdx.x;          // one workgroup per image

    // -------- zero histogram --------
    for (int i = tid; i < NB; i += THREADS) { s_cp[i] = 0ull; s_fx[i] = 0ull; }
    __syncthreads();

    // -------- pass 1: stream image, BCE partial + LDS histogram --------
    const float4* p4 = (const float4*)(pred + (size_t)img * P_IMG);
    const float4* t4 = (const float4*)(targ + (size_t)img * P_IMG);
    float bce = 0.0f;

    for (int i = tid; i < P_IMG / 4; i += THREADS) {
        // speculative prefetch of the next tile (global_prefetch_b8)
        __builtin_prefetch((const void*)(p4 + i + THREADS), 0, 0);
        __builtin_prefetch((const void*)(t4 + i + THREADS), 0, 0);
        float4 xv = p4[i];
        float4 tv = t4[i];
        const float xs[4] = {xv.x, xv.y, xv.z, xv.w};
        const float ts[4] = {tv.x, tv.y, tv.z, tv.w};
        #pragma unroll
        for (int c = 0; c < 4; ++c) {
            float x = xs[c], t = ts[c];
            // numerically stable BCE-with-logits term
            bce += fmaxf(x, 0.0f) - x * t + log1pf(expf(-fabsf(x)));
            // hinge error; bucket 0 = largest e  (descending order scan)
            float e = fmaf(-x, 2.0f * t - 1.0f, 1.0f);
            int b = (int)((EMAX - e) * BPU);
            b = b < 0 ? 0 : (b > NB - 1 ? NB - 1 : b);
            unsigned long long inc =
                1ull | ((unsigned long long)(t > 0.5f) << 32);
            atomicAdd(&s_cp[b], inc);                       // ds_add_u64
            if (e > 0.0f) {
                atomicAdd(&s_fx[b], (unsigned long long)(e * FXS));
            }
        }
    }
    __syncthreads();

    // -------- pass 2: scan 16384 buckets (16 per thread) --------
    float c[16], q[16], cumC[16], cumQ[16];
    float tC = 0.0f, tQ = 0.0f;
    const int base = tid * 16;
    #pragma unroll
    for (int k = 0; k < 16; ++k) {
        unsigned long long cp = s_cp[base + k];
        c[k] = (float)(unsigned int)(cp & 0xffffffffull);
        q[k] = (float)(unsigned int)(cp >> 32);
        tC += c[k]; cumC[k] = tC;
        tQ += q[k]; cumQ[k] = tQ;
    }

    // wave32 inclusive scan of thread totals
    float incC = tC, incQ = tQ;
    #pragma unroll
    for (int d = 1; d < 32; d <<= 1) {
        float oC = __shfl_up(incC, d);
        float oQ = __shfl_up(incQ, d);
        if (lane >= d) { incC += oC; incQ += oQ; }
    }
    if (lane == 31) { s_waveC[wave] = incC; s_waveQ[wave] = incQ; }
    __syncthreads();

    // -------- cross-wave prefix scan on the tensor pipe (wave 0 only) ------
    // P = T x S,  T = 16x16 lower-triangular ones, chained as four 16x16x4
    // f32 WMMAs.  S columns: 0: cnt[0..15], 1: cnt[16..31],
    //                        2: pos[0..15], 3: pos[16..31].
    if (wave == 0) {                      // all 32 lanes active -> EXEC ok
        const int half = lane >> 4;       // A/B layout: lanes 16-31 hold K+2
        const int mn   = lane & 15;       // A row / B,D column
        v8f acc = {};
        #pragma unroll
        for (int k = 0; k < 4; ++k) {
            const int r0 = 4 * k + (half ? 2 : 0);
            const int r1 = r0 + 1;
            v2f a, b;
            a.x = (r0 <= mn) ? 1.0f : 0.0f;   // T[mn][r0]
            a.y = (r1 <= mn) ? 1.0f : 0.0f;   // T[mn][r1]
            // S[r][mn]
            b.x = (mn == 0) ? s_waveC[r0] : (mn == 1) ? s_waveC[16 + r0]
                : (mn == 2) ? s_waveQ[r0] : (mn == 3) ? s_waveQ[16 + r0] : 0.0f;
            b.y = (mn == 0) ? s_waveC[r1] : (mn == 1) ? s_waveC[16 + r1]
                : (mn == 2) ? s_waveQ[r1] : (mn == 3) ? s_waveQ[16 + r1] : 0.0f;
            acc = __builtin_amdgcn_wmma_f32_16x16x4_f32(
                false, a, false, b, (short)0, acc, false, false);
        }
        // D layout: lane L holds P[j + (L<16?0:8)][L&15] in acc[j].
        float totC0 = __shfl(acc[7], 16);   // P[15][0] = sum cnt[0..15]
        float totQ0 = __shfl(acc[7], 18);   // P[15][2] = sum pos[0..15]
        if (mn < 4) {
            #pragma unroll
            for (int j = 0; j < 8; ++j) {
                int m = j + (half ? 8 : 0);
                float v = acc[j];
                if      (mn == 0) s_inclC[m]      = v;
                else if (mn == 1) s_inclC[16 + m] = v + totC0;
                else if (mn == 2) s_inclQ[m]      = v;
                else              s_inclQ[16 + m] = v + totQ0;
            }
        }
    }
    __syncthreads();

    const float gts      = s_inclQ[31];                    // total positives
    const float thrExclC = (s_inclC[wave] - s_waveC[wave]) + (incC - tC);
    const float thrExclQ = (s_inclQ[wave] - s_waveQ[wave]) + (incQ - tQ);

    // -------- per-bucket tie-group Lovasz contribution --------
    float lov = 0.0f;
    #pragma unroll
    for (int k = 0; k < 16; ++k) {
        float n = c[k];
        if (n > 0.0f) {
            float np   = q[k];
            float a    = thrExclC + cumC[k] - n;   // elements ranked before
            float p    = thrExclQ + cumQ[k] - np;  // positives ranked before
            float Jend = jacc(a + n, p + np, gts);
            float Jpre = (a > 0.0f) ? jacc(a, p, gts) : 0.0f;
            float sr   = (float)s_fx[base + k] * FXS_INV;  // sum relu(e)
            lov += sr * (Jend - Jpre) / n;         // mean-grad pairing
        }
    }

    // -------- block reductions (deterministic tree) --------
    #pragma unroll
    for (int d = 16; d >= 1; d >>= 1) {
        lov += __shfl_xor(lov, d);
        bce += __shfl_xor(bce, d);
    }
    __syncthreads();                       // done reading s_waveC/Q: reuse
    if (lane == 0) { s_waveC[wave] = lov; s_waveQ[wave] = bce; }
    __syncthreads();
    if (wave == 0) {
        float l = s_waveC[lane];
        float b = s_waveQ[lane];
        #pragma unroll
        for (int d = 16; d >= 1; d >>= 1) {
            l += __shfl_xor(l, d);
            b += __shfl_xor(b, d);
        }
        if (lane == 0) { ws[img] = l; ws[B_IMG + img] = b; }
    }
}

__global__ void finalize_kernel(const float* __restrict__ ws,
                                float* __restrict__ out) {
    const int tid  = threadIdx.x;          // 128 threads = 4 waves
    const int lane = tid & 31;
    const int wave = tid >> 5;
    __shared__ float sl[4], sb[4];
    float l = ws[tid];
    float b = ws[B_IMG + tid];
    #pragma unroll
    for (int d = 16; d >= 1; d >>= 1) {
        l += __shfl_xor(l, d);
        b += __shfl_xor(b, d);
    }
    if (lane == 0) { sl[wave] = l; sb[wave] = b; }
    __syncthreads();
    if (tid == 0) {
        float ls = sl[0] + sl[1] + sl[2] + sl[3];
        float bs = sb[0] + sb[1] + sb[2] + sb[3];
        out[0] = bs / (float)N_TOT + ls / (float)B_IMG;
    }
}

extern "C" void kernel_launch(void* const* d_in, const int* in_sizes, int n_in,
                              void* d_out, int out_size, void* d_ws, size_t ws_size,
                              hipStream_t stream) {
    const float* pred = (const float*)d_in[0];
    const float* targ = (const float*)d_in[1];
    float* out = (float*)d_out;
    float* ws  = (float*)d_ws;   // 256 floats: per-image lovasz + bce partials

    lovasz_bce_kernel<<<B_IMG, THREADS, 0, stream>>>(pred, targ, ws);
    finalize_kernel<<<1, 128, 0, stream>>>(ws, out);
}